// GNNGuard_14585708937909
// MI455X (gfx1250) — compile-verified
//
#include <hip/hip_runtime.h>
#include <math.h>

typedef float v2f __attribute__((ext_vector_type(2)));
typedef float v8f __attribute__((ext_vector_type(8)));

#define D_FEAT 128
#define SIM_THRESH 0.1f

// ---------------------------------------------------------------------------
// Kernel A: per-node inverse L2 norm (one wave32 per node, b128 loads,
// shfl_xor tree reduce) + zero the row-sum accumulator.
// ---------------------------------------------------------------------------
__global__ void gnn_norms_kernel(const float* __restrict__ feat,
                                 float* __restrict__ invn,
                                 float* __restrict__ row_sum,
                                 int n_node) {
  int tid  = blockIdx.x * blockDim.x + threadIdx.x;
  if (tid < n_node) row_sum[tid] = 0.0f;

  int wave = tid >> 5;                 // one wave per feature row
  int lane = threadIdx.x & 31;
  if (wave < n_node) {
    const float4* fp = (const float4*)(feat + (size_t)wave * D_FEAT) + lane;
    float4 v = *fp;                    // global_load_b128
    float s = v.x * v.x + v.y * v.y + v.z * v.z + v.w * v.w;
    s += __shfl_xor(s, 16, 32);
    s += __shfl_xor(s,  8, 32);
    s += __shfl_xor(s,  4, 32);
    s += __shfl_xor(s,  2, 32);
    s += __shfl_xor(s,  1, 32);
    if (lane == 0) {
      float nrm = fmaxf(sqrtf(s), 1e-12f);
      invn[wave] = 1.0f / nrm;
    }
  }
}

// ---------------------------------------------------------------------------
// Kernel B: 16 edges per wave. D = A(16xK rows fn[row]) x B(Kx16 cols fn[col])
// via a chain of V_WMMA_F32_16X16X4_F32; the diagonal of D gives the 16 edge
// dot products in full f32. Feature gathers (25.6 MB working set) live in L2.
// ---------------------------------------------------------------------------
__global__ void gnn_edge_sim_wmma_kernel(const int* __restrict__ ei,      // [2*E]
                                         const float* __restrict__ feat,
                                         const float* __restrict__ invn,
                                         float* __restrict__ sim,
                                         float* __restrict__ row_sum,
                                         int n_edge, int n_tiles) {
  int gtid = blockIdx.x * blockDim.x + threadIdx.x;
  int tile = gtid >> 5;
  if (tile >= n_tiles) return;         // wave-uniform: EXEC stays all-ones

  int l  = threadIdx.x & 31;
  int m  = l & 15;                     // edge slot within tile
  int hi = l >> 4;                     // K half select (K={0,1} vs K={2,3})
  long base   = (long)tile * 16;
  long e_load = base + m;
  if (e_load > (long)n_edge - 1) e_load = (long)n_edge - 1;  // clamp tail dups

  int r = ei[e_load];                  // row index
  int c = ei[(size_t)n_edge + e_load]; // col index

  // Identical per-lane addressing for A (rows) and B (cols):
  // lane holds (K = 2*hi, 2*hi+1) of feature vector m -> one float2 per step.
  const float* ap = feat + (size_t)r * D_FEAT + 2 * hi;
  const float* bp = feat + (size_t)c * D_FEAT + 2 * hi;

  v8f acc = {};
#pragma unroll
  for (int k = 0; k < D_FEAT; k += 4) {
    v2f a = *(const v2f*)(ap + k);     // global_load_b64
    v2f b = *(const v2f*)(bp + k);     // global_load_b64
    acc = __builtin_amdgcn_wmma_f32_16x16x4_f32(
        /*neg_a=*/false, a, /*neg_b=*/false, b,
        /*c_mod=*/(short)0, acc, /*reuse_a=*/false, /*reuse_b=*/false);
  }

  // Diagonal element M lives at (VGPR = M&7, lane = M<8 ? M : M+16).
  int sub = l & 7;
  float d = acc[0];
  if (sub == 1) d = acc[1];
  if (sub == 2) d = acc[2];
  if (sub == 3) d = acc[3];
  if (sub == 4) d = acc[4];
  if (sub == 5) d = acc[5];
  if (sub == 6) d = acc[6];
  if (sub == 7) d = acc[7];

  bool active = (l < 8) || (l >= 16 && l < 24);
  int  mm     = (l < 8) ? l : (l - 8); // output edge slot for this lane
  long e      = base + mm;
  if (active && e < (long)n_edge) {
    int rr = ei[e];
    int cc = ei[(size_t)n_edge + e];
    float s = d * invn[rr] * invn[cc];           // cosine similarity (f32 exact)
    s = (s < SIM_THRESH) ? 0.0f : s;             // threshold
    sim[e] = s;
    if (s > 0.0f) atomicAdd(&row_sum[rr], s);    // |s| == s after threshold
  }
}

// ---------------------------------------------------------------------------
// Kernel C: out[e] = max(g*w[e] + (1-g)*exp(sim[e]/denom[row[e]]), 0)
// (self-loops excluded by construction -> lambda/degree path is dead code)
// ---------------------------------------------------------------------------
__global__ void gnn_finalize_kernel(const int* __restrict__ ei,
                                    const float* __restrict__ ew,
                                    const float* __restrict__ gate,
                                    const float* __restrict__ sim,
                                    const float* __restrict__ row_sum,
                                    float* __restrict__ out,
                                    int n_edge) {
  int e = blockIdx.x * blockDim.x + threadIdx.x;
  if (e >= n_edge) return;
  float g  = gate[0];
  int   r  = ei[e];
  float rs = row_sum[r];
  float denom = (rs > 0.0f) ? rs : 1.0f;
  float att = expf(sim[e] / denom);
  float o = g * ew[e] + (1.0f - g) * att;
  out[e] = fmaxf(o, 0.0f);
}

// ---------------------------------------------------------------------------
// Launch: 3 stream-ordered kernels; scratch layout in d_ws (floats):
//   [invn: n_node][row_sum: n_node][sim: n_edge]  (~2.9 MB total)
// ---------------------------------------------------------------------------
extern "C" void kernel_launch(void* const* d_in, const int* in_sizes, int n_in,
                              void* d_out, int out_size, void* d_ws, size_t ws_size,
                              hipStream_t stream) {
  const int*   ei   = (const int*)d_in[0];    // edge_index [2, E]
  const float* ew   = (const float*)d_in[1];  // edge_weight [E]
  const float* feat = (const float*)d_in[2];  // features [N, 128]
  const float* gate = (const float*)d_in[3];  // gate [1]

  int n_edge = in_sizes[1];
  int n_node = in_sizes[2] / D_FEAT;

  float* wsf     = (float*)d_ws;
  int    padN    = (n_node + 127) & ~127;     // 128-elem aligned segments
  float* invn    = wsf;
  float* row_sum = wsf + padN;
  float* simbuf  = wsf + 2 * (size_t)padN;
  float* out     = (float*)d_out;

  // A: one wave per node (8 waves / 256-thread block)
  int blocksA = (n_node + 7) / 8;
  gnn_norms_kernel<<<blocksA, 256, 0, stream>>>(feat, invn, row_sum, n_node);

  // B: one 16-edge tile per wave
  int n_tiles = (n_edge + 15) / 16;
  int blocksB = (n_tiles + 7) / 8;
  gnn_edge_sim_wmma_kernel<<<blocksB, 256, 0, stream>>>(ei, feat, invn, simbuf,
                                                        row_sum, n_edge, n_tiles);

  // C: elementwise finalize
  int blocksC = (n_edge + 255) / 256;
  gnn_finalize_kernel<<<blocksC, 256, 0, stream>>>(ei, ew, gate, simbuf,
                                                   row_sum, out, n_edge);
}